// Gridding_79525614452761
// MI455X (gfx1250) — compile-verified
//
#include <hip/hip_runtime.h>
#include <stdint.h>

#define GLOBAL_AS __attribute__((address_space(1)))
#define LOCAL_AS  __attribute__((address_space(3)))

constexpr int kRegions = 17;
constexpr int kCh      = 7;
constexpr int kRowsG   = 82;
constexpr int kColsG   = 67;
constexpr int kCells   = 3000;
constexpr int kBatch   = 2048;
constexpr int kPos     = kRowsG * kColsG;   // 5494 grid positions
constexpr int kSlab    = kPos * kCh;        // 38458 floats per batch (8B aligned stride)
constexpr int kXRow    = kRegions * kCh;    // 119 floats per batch row of x
constexpr int kThreads = 256;               // 8 wave32s

typedef float v2f __attribute__((ext_vector_type(2)));
typedef int   v4i __attribute__((vector_size(16)));   // matches builtin's b128 pointee

// ---- CDNA5 async global->LDS DMA (ASYNCcnt path), builtin with asm fallback ----
__device__ __forceinline__ uint32_t lds_offset(void* p) {
    // generic -> AS(3) cast, then ptrtoint: yields byte offset within LDS
    return (uint32_t)(size_t)(LOCAL_AS void*)p;
}

__device__ __forceinline__ void async_copy_b128(const void* g, void* l) {
#if __has_builtin(__builtin_amdgcn_global_load_async_to_lds_b128)
    __builtin_amdgcn_global_load_async_to_lds_b128(
        (GLOBAL_AS v4i*)g, (LOCAL_AS v4i*)l, 0, 0);
#else
    asm volatile("global_load_async_to_lds_b128 %0, %1, off"
                 :: "v"(lds_offset(l)), "v"((uint64_t)(size_t)g) : "memory");
#endif
}

__device__ __forceinline__ void async_copy_b32(const void* g, void* l) {
#if __has_builtin(__builtin_amdgcn_global_load_async_to_lds_b32)
    __builtin_amdgcn_global_load_async_to_lds_b32(
        (GLOBAL_AS int*)g, (LOCAL_AS int*)l, 0, 0);
#else
    asm volatile("global_load_async_to_lds_b32 %0, %1, off"
                 :: "v"(lds_offset(l)), "v"((uint64_t)(size_t)g) : "memory");
#endif
}

__device__ __forceinline__ void wait_async_zero() {
#if __has_builtin(__builtin_amdgcn_s_wait_asynccnt)
    __builtin_amdgcn_s_wait_asynccnt(0);
#else
    asm volatile("s_wait_asynccnt 0x0" ::: "memory");
#endif
}

// One block per batch: stage indices + x-row into LDS via async DMA, build the
// inverse scatter table, then stream the 150 KB output slab with NT b64 stores.
__global__ __launch_bounds__(kThreads) void gridding_scatter_kernel(
    const float* __restrict__ x,          // (2048, 119)
    const int*   __restrict__ cell_lin,   // (3000,)
    const int*   __restrict__ region_ids, // (3000,) in [0,17)
    float*       __restrict__ out)        // (2048, 82, 67, 7)
{
    __shared__ int   s_cl[kCells];    // staged cell_lin           (12000 B)
    __shared__ int   s_rid[kCells];   // staged region_ids         (12000 B)
    __shared__ int   s_src[kPos];     // pos -> x element offset   (21976 B)
    __shared__ float s_x[kXRow];      // this batch's x row        (476 B)

    const int tid = threadIdx.x;
    const int b   = blockIdx.x;

    // 1) kick off async DMA of the index arrays (750 x b128 each) and x row
    constexpr int n128 = (kCells * 4) / 16;  // 750
    for (int i = tid; i < n128; i += kThreads) {
        async_copy_b128((const char*)cell_lin   + i * 16, (char*)s_cl  + i * 16);
        async_copy_b128((const char*)region_ids + i * 16, (char*)s_rid + i * 16);
    }
    const float* xrow = x + (size_t)b * kXRow;
    for (int i = tid; i < kXRow; i += kThreads)
        async_copy_b32(xrow + i, s_x + i);

    // 2) overlap: initialize scatter table while DMA is in flight
    for (int i = tid; i < kPos; i += kThreads) s_src[i] = -1;

    wait_async_zero();
    __syncthreads();

    // 3) build inverse mapping: grid position -> offset into x row
    for (int c = tid; c < kCells; c += kThreads)
        s_src[s_cl[c]] = s_rid[c] * kCh;
    __syncthreads();

    // 4) stream the slab: 2 floats/thread/iter, 8B-aligned non-temporal stores
    v2f* outb = (v2f*)(out + (size_t)b * kSlab);   // base: b*153832 B, 8B aligned
    constexpr int nPairs = kSlab / 2;              // 19229 (kSlab is even)
    for (int i = tid; i < nPairs; i += kThreads) {
        const unsigned p0 = 2u * (unsigned)i;
        const unsigned p1 = p0 + 1u;
        const unsigned cell0 = p0 / 7u, ch0 = p0 - cell0 * 7u;
        const unsigned cell1 = p1 / 7u, ch1 = p1 - cell1 * 7u;
        const int so0 = s_src[cell0];
        const int so1 = s_src[cell1];
        v2f v;
        v.x = (so0 >= 0) ? s_x[so0 + (int)ch0] : 0.0f;
        v.y = (so1 >= 0) ? s_x[so1 + (int)ch1] : 0.0f;
        __builtin_nontemporal_store(v, &outb[i]);
    }
}

extern "C" void kernel_launch(void* const* d_in, const int* in_sizes, int n_in,
                              void* d_out, int out_size, void* d_ws, size_t ws_size,
                              hipStream_t stream) {
    (void)in_sizes; (void)n_in; (void)out_size; (void)d_ws; (void)ws_size;
    const float* x          = (const float*)d_in[0];
    const int*   cell_lin   = (const int*)d_in[1];
    const int*   region_ids = (const int*)d_in[2];
    float*       out        = (float*)d_out;

    gridding_scatter_kernel<<<dim3(kBatch), dim3(kThreads), 0, stream>>>(
        x, cell_lin, region_ids, out);
}